// Fume3dLayer_48799418417413
// MI455X (gfx1250) — compile-verified
//
#include <hip/hip_runtime.h>
#include <stdint.h>

#ifndef __has_builtin
#define __has_builtin(x) 0
#endif

typedef uint32_t u32x4 __attribute__((ext_vector_type(4)));
typedef int      i32x4 __attribute__((ext_vector_type(4)));
typedef int      i32x8 __attribute__((ext_vector_type(8)));
typedef float    v2f   __attribute__((ext_vector_type(2)));
typedef float    v8f   __attribute__((ext_vector_type(8)));

#define IMG_H 256
#define IMG_W 256
#define N_IMG 32                      // B*C = 8*4
#define X_STRIPS 16                   // 16 x-columns per block
#define LDS_STRIDE 257                // dwords: +1 pad/row -> bank = (row+u) % 64

#if __has_builtin(__builtin_amdgcn_rcpf)
__device__ __forceinline__ float fast_rcp(float x)  { return __builtin_amdgcn_rcpf(x); }
#else
__device__ __forceinline__ float fast_rcp(float x)  { return 1.0f / x; }
#endif
#if __has_builtin(__builtin_amdgcn_sqrtf)
__device__ __forceinline__ float fast_sqrt(float x) { return __builtin_amdgcn_sqrtf(x); }
#else
__device__ __forceinline__ float fast_sqrt(float x) { return sqrtf(x); }
#endif

// ---------------------------------------------------------------------------
// Epipolar line-integral ("fume translate") kernel.
//  * TDM stages one full (b,c) image (256x256 f32) into LDS with hardware
//    row padding (stride 257 dwords) for bank de-confliction.
//  * V_WMMA_F32_16X16X4_F32 computes the per-pixel epipolar line coefficients
//    for a 16(y) x 16(x) tile per wave:  D = A(16x4) x B(4x16),
//    A row m = [y_m, 1, 0, 0],  B col n = [F_y, F_x*x_n + F_1, 0, 0]
//    => D[m][n] = F_x*x_n + F_y*y_m + F_1.  The C/D lane layout hands each
//    lane its own coefficients (VGPR r = y_base + 8*(lane>>4) + r, N = lane&15),
//    so no cross-lane shuffles are needed.
//  * Each lane then integrates 8 epipolar lines out of LDS, with v_rcp/v_sqrt
//    hardware approximations for the per-line setup.
// ---------------------------------------------------------------------------
__global__ void __launch_bounds__(256)
fume_epipolar_kernel(const float* __restrict__ view,
                     const float* __restrict__ Fm,
                     float* __restrict__ out)
{
    extern __shared__ float smem[];   // IMG_H * LDS_STRIDE floats (padded rows)

    const int img = blockIdx.x;       // 0..31  (b*4 + c)
    const int xs  = blockIdx.y;       // 0..15  (16-wide x strip)
    const float* __restrict__ src = view + (size_t)img * (IMG_H * IMG_W);

#if __has_builtin(__builtin_amdgcn_tensor_load_to_lds)
    // --- TDM: DMA the whole image into LDS, hardware-padding each 256-dword
    //     row with 1 dword so the LDS row stride is 257 (bank de-conflict).
    if (threadIdx.x < 32) {           // one wave issues the tensor op
        const uint64_t ga = (uint64_t)(uintptr_t)src;
        const uint32_t la = (uint32_t)(uintptr_t)smem;   // LDS byte offset

        u32x4 g0;
        g0[0] = 1u;                                       // count=1, user mode
        g0[1] = la;                                       // lds_addr
        g0[2] = (uint32_t)ga;                             // global_addr[31:0]
        g0[3] = (uint32_t)((ga >> 32) & 0x01FFFFFFu)      // global_addr[56:32]
              | (2u << 30);                               // type = 2 ("image")

        i32x8 g1;
        g1[0] = (int)((2u << 16)      // data_size = 4 bytes
                    | (1u << 20)      // pad_enable
                    | (7u << 22)      // pad_interval: every 256 dwords
                    | (0u << 25));    // pad_amount: 1 dword
        g1[1] = (int)(((uint32_t)IMG_W & 0xFFFFu) << 16); // tensor_dim0 lo16
        g1[2] = (int)(((uint32_t)IMG_H & 0xFFFFu) << 16); // dim0 hi=0 | tensor_dim1 lo16
        g1[3] = (int)(((uint32_t)IMG_W) << 16);           // dim1 hi=0 | tile_dim0
        g1[4] = (int)(IMG_H);                             // tile_dim1 | tile_dim2=0
        g1[5] = (int)(IMG_W);                             // tensor_dim0_stride[31:0]
        g1[6] = 0;                                        // stride0 hi | stride1 lo16 (=0)
        g1[7] = 1;                                        // stride1[47:16]=1 -> 65536
        i32x4 gz = {0, 0, 0, 0};
#if __clang_major__ >= 23
        i32x8 gz8 = {0, 0, 0, 0, 0, 0, 0, 0};
        __builtin_amdgcn_tensor_load_to_lds(g0, g1, gz, gz, gz8, 0);
#else
        __builtin_amdgcn_tensor_load_to_lds(g0, g1, gz, gz, 0);
#endif
        __builtin_amdgcn_s_wait_tensorcnt(0);
    }
#else
    // --- Fallback: cooperative b128 loads, manual row padding.
    for (int idx = threadIdx.x; idx < (IMG_H * IMG_W) / 4; idx += blockDim.x) {
        const int row = idx >> 6;                // 64 float4 per row
        const int q   = (idx & 63) << 2;
        const float4 v4 = reinterpret_cast<const float4*>(src)[idx];
        float* d = &smem[row * LDS_STRIDE + q];
        d[0] = v4.x; d[1] = v4.y; d[2] = v4.z; d[3] = v4.w;
    }
#endif
    __syncthreads();

    // Fundamental matrix (row-major 3x3), uniform across the block.
    // Coefficients (reference): a = F[0,0]x + F[1,0]y + F[2,0], etc.
    const float Fa_x = Fm[0], Fb_x = Fm[1], Fc_x = Fm[2];
    const float Fa_y = Fm[3], Fb_y = Fm[4], Fc_y = Fm[5];
    const float Fa_1 = Fm[6], Fb_1 = Fm[7], Fc_1 = Fm[8];

    const int lane = threadIdx.x & 31;
    const int wave = threadIdx.x >> 5;           // 8 waves: y sub-tiles
    const int half = lane >> 4;                  // 0: M=0..7, 1: M=8..15
    const int x    = xs * 16 + (lane & 15);      // N = lane & 15
    const float xf = (float)x;

#if __has_builtin(__builtin_amdgcn_wmma_f32_16x16x4_f32)
    // B operands depend only on x (K=0 row: y-coefficient; K=1 row: affine part;
    // K=2,3 rows — held by lanes 16..31 — are zero).
    v2f Ba, Bb, Bc;
    Ba[0] = half ? 0.0f : Fa_y;  Ba[1] = half ? 0.0f : fmaf(Fa_x, xf, Fa_1);
    Bb[0] = half ? 0.0f : Fb_y;  Bb[1] = half ? 0.0f : fmaf(Fb_x, xf, Fb_1);
    Bc[0] = half ? 0.0f : Fc_y;  Bc[1] = half ? 0.0f : fmaf(Fc_x, xf, Fc_1);
#endif

    for (int t = 0; t < 2; ++t) {                // two 16x16 tiles per wave
        const int y_base = t * 128 + wave * 16;

        float ca[8], cb[8], cc[8];
#if __has_builtin(__builtin_amdgcn_wmma_f32_16x16x4_f32)
        // A row m = [y_m, 1, 0, 0]; lanes 0..15 hold K=0,1; lanes 16..31 K=2,3.
        v2f A;
        A[0] = half ? 0.0f : (float)(y_base + (lane & 15));
        A[1] = half ? 0.0f : 1.0f;
        const v8f z = {};
        v8f da = __builtin_amdgcn_wmma_f32_16x16x4_f32(false, A, false, Ba,
                                                       (short)0, z, false, false);
        v8f db = __builtin_amdgcn_wmma_f32_16x16x4_f32(false, A, false, Bb,
                                                       (short)0, z, false, false);
        v8f dc = __builtin_amdgcn_wmma_f32_16x16x4_f32(false, A, false, Bc,
                                                       (short)0, z, false, false);
#pragma unroll
        for (int r = 0; r < 8; ++r) { ca[r] = da[r]; cb[r] = db[r]; cc[r] = dc[r]; }
#else
#pragma unroll
        for (int r = 0; r < 8; ++r) {
            const float yf = (float)(y_base + half * 8 + r);
            ca[r] = fmaf(Fa_x, xf, fmaf(Fa_y, yf, Fa_1));
            cb[r] = fmaf(Fb_x, xf, fmaf(Fb_y, yf, Fb_1));
            cc[r] = fmaf(Fc_x, xf, fmaf(Fc_y, yf, Fc_1));
        }
#endif

        for (int r = 0; r < 8; ++r) {
            const int y = y_base + half * 8 + r;
            float a = ca[r], b = cb[r], c = cc[r];
            // Hardware-rate rcp/sqrt: replaces ~40-op IEEE div/sqrt expansions.
            const float rn = fast_rcp(fast_sqrt(fmaf(a, a, b * b)) + 1e-12f);
            a *= rn; b *= rn; c *= rn;
            const bool  ok = fabsf(b) > 1e-6f;
            const float bs = ok ? b : 1.0f;
            const float rb = -fast_rcp(bs);      // divisions hoisted out of loop
            const float dv = a * rb;             // v(u) = dv*u + v0
            const float v0 = c * rb;

            float acc = 0.0f;
#pragma unroll 4
            for (int u = 0; u < IMG_W; ++u) {
                const float v  = fmaf(dv, (float)u, v0);
                const float fl = floorf(v);
                const float f  = v - fl;
                const int   i0 = (int)fl;
                const int   i1 = i0 + 1;
                const float w0 = ((unsigned)i0 < (unsigned)IMG_H) ? (1.0f - f) : 0.0f;
                const float w1 = ((unsigned)i1 < (unsigned)IMG_H) ? f : 0.0f;
                const int   c0 = i0 < 0 ? 0 : (i0 > IMG_H - 1 ? IMG_H - 1 : i0);
                const int   c1 = i1 < 0 ? 0 : (i1 > IMG_H - 1 ? IMG_H - 1 : i1);
                const float s0 = smem[c0 * LDS_STRIDE + u];
                const float s1 = smem[c1 * LDS_STRIDE + u];
                acc = fmaf(s0, w0, acc);
                acc = fmaf(s1, w1, acc);
            }
            out[(size_t)img * (IMG_H * IMG_W) + (size_t)y * IMG_W + x] =
                ok ? acc : 0.0f;
        }
    }
}

// ---------------------------------------------------------------------------
extern "C" void kernel_launch(void* const* d_in, const int* in_sizes, int n_in,
                              void* d_out, int out_size, void* d_ws, size_t ws_size,
                              hipStream_t stream) {
    (void)in_sizes; (void)n_in; (void)d_ws; (void)ws_size; (void)out_size;
    const float* view = (const float*)d_in[0];   // (8,4,256,256) f32
    const float* Fm   = (const float*)d_in[1];   // (3,3) f32
    float*       out  = (float*)d_out;           // (8,4,256,256) f32

    const dim3 grid(N_IMG, X_STRIPS);
    const dim3 block(256);
    const size_t shmem = (size_t)IMG_H * LDS_STRIDE * sizeof(float); // ~257 KB
    fume_epipolar_kernel<<<grid, block, shmem, stream>>>(view, Fm, out);
}